// ATSSPostProcessor_56573309224788
// MI455X (gfx1250) — compile-verified
//
#include <hip/hip_runtime.h>
#include <hip/hip_bf16.h>
#include <math.h>

// ---------------- problem constants (match reference) ----------------
#define NIMG   8
#define NCLS   80
#define HH     160
#define WW     160
#define HWSZ   (HH * WW)            // 25600 = L
#define CHW    (NCLS * HWSZ)        // 2,048,000 scores per image
#define TOPK   1000
#define PADK   1024
#define POSTK  100
#define TILE   8192                 // elements staged per block (32 KB LDS)
#define TPB    256
#define TILES_PER_IMG (CHW / TILE)  // 250 exactly
#define IMGM1  1279.0f              // IMG_SIZE - 1
#define XCLIP  4.135166556742356f   // log(1000/16)
#define NMS_T  0.6f

// ws layout (bytes)
#define HIST_OFF   0                      // u32 [8][2048]
#define PFX_OFF    65536                  // u32 [8]
#define NEED_OFF   65568                  // u32 [8]
#define CNTHI_OFF  65600                  // u32 [8]
#define CNTEQ_OFF  65632                  // u32 [8]
#define PAIRS_OFF  65664                  // u64 [8][1024]
#define ZERO_BYTES 131200                 // memset range per launch
#define SCTR_OFF   131200                 // f32 [8][25600]  (819200 B)
#define KEYS_OFF   950400                 // u32 [8][2048000] (65,536,000 B)
#define WS_BIG     (950400ull + (unsigned long long)NIMG * CHW * 4ull)

typedef __attribute__((ext_vector_type(4))) unsigned int v4u;
typedef __attribute__((ext_vector_type(8))) int          v8i;
typedef __attribute__((ext_vector_type(4))) int          v4i;

// monotonic float->uint key (descending top-k = largest keys)
__device__ __forceinline__ unsigned keyOf(float s) {
    unsigned u = __float_as_uint(s);
    return ((int)u < 0) ? ~u : (u | 0x80000000u);
}
__device__ __forceinline__ float valOf(unsigned k) {
    unsigned u = (k & 0x80000000u) ? (k & 0x7FFFFFFFu) : ~k;
    return __uint_as_float(u);
}
__device__ __forceinline__ float sigm(float x) { return 1.0f / (1.0f + expf(-x)); }

// ---- stage TILE 4-byte elements from global into LDS (offset 0) via TDM ----
// D# per CDNA5 ISA ch.8: group0 = {count=1, lds_addr, global_addr, type=2},
// group1 = {data_size=4B, tensor_dim0=TILE, tile_dim0=TILE (1-D tile)}.
// This toolchain's builtin is the 6-arg form:
//   (uint32x4 g0, int32x8 g1, int32x4 g2, int32x4 g3, int32x8 extra, i32 cpol)
__device__ __forceinline__ void stage_tile(const void* __restrict__ gsrc, void* tile) {
#if defined(__HIP_DEVICE_COMPILE__) && __has_builtin(__builtin_amdgcn_tensor_load_to_lds)
    (void)tile; // tile is the first/largest LDS object -> byte offset 0
    if (threadIdx.x < 32u) { // one wave issues; TDM ignores EXEC
        unsigned long long ga = (unsigned long long)(__SIZE_TYPE__)gsrc;
        v4u g0 = { 1u,                                        // count=1 (valid user D#)
                   0u,                                        // lds_addr = 0
                   (unsigned)(ga & 0xFFFFFFFFull),
                   (unsigned)((ga >> 32) & 0x01FFFFFFull) | (2u << 30) }; // type=2
        v8i g1 = { (int)(2u << 16),                           // data_size=2 (4 bytes)
                   (int)((TILE & 0xFFFFu) << 16),             // tensor_dim0[15:0]
                   (int)(((TILE >> 16) & 0xFFFFu) | (1u << 16)), // tensor_dim0 hi, tensor_dim1=1
                   (int)(TILE << 16),                         // tile_dim0 = TILE
                   0,                                         // tile_dim1=0, tile_dim2=0
                   (int)TILE,                                 // tensor_dim0_stride lo
                   0, 0 };
        v4i gz = { 0, 0, 0, 0 };
        v8i z8 = { 0, 0, 0, 0, 0, 0, 0, 0 };
        __builtin_amdgcn_tensor_load_to_lds(g0, g1, gz, gz, z8, 0);
        __builtin_amdgcn_s_wait_tensorcnt(0);
    }
    __asm__ volatile("" ::: "memory");
#else
    const float4* g4 = (const float4*)gsrc;
    float4*       t4 = (float4*)tile;
    for (int k = threadIdx.x; k < TILE / 4; k += TPB) t4[k] = g4[k];
#endif
    __syncthreads();
}

// ---------------- kernels ----------------

__global__ __launch_bounds__(256) void k_ctr(const float* __restrict__ ctr, float* __restrict__ sctr) {
    int i = blockIdx.x * 256 + threadIdx.x;
    if (i < NIMG * HWSZ) sctr[i] = sigm(ctr[i]);
}

// ---- pass 1 (keys-cache variant): compute keys once, store them, histogram bits[31:21]
__global__ __launch_bounds__(TPB) void k_hist_store(const float* __restrict__ cls,
                                                    const float* __restrict__ sctr,
                                                    unsigned* __restrict__ keys,
                                                    unsigned* __restrict__ hist) {
    __shared__ __align__(16) float tile[TILE];
    __shared__ unsigned lhist[2048];
    const int n  = blockIdx.y;
    const int t0 = blockIdx.x * TILE;
    for (unsigned b = threadIdx.x; b < 2048u; b += TPB) lhist[b] = 0u;
    stage_tile(cls + (size_t)n * CHW + t0, tile);   // includes __syncthreads()
    const float* ctrb = sctr + n * HWSZ;
    unsigned* kb = keys + (size_t)n * CHW + t0;
    for (int k = 0; k < TILE / TPB; ++k) {
        int   t = threadIdx.x + k * TPB;
        int   l = (t0 + t) % HWSZ;                  // memory order: c-major
        float p = sigm(tile[t]);
        float s = (p > 0.05f) ? p * ctrb[l] : -INFINITY;  // cand mask pre-centerness
        unsigned key = keyOf(s);
        kb[t] = key;
        atomicAdd(&lhist[key >> 21], 1u);
    }
    __syncthreads();
    for (unsigned b = threadIdx.x; b < 2048u; b += TPB) {
        unsigned c = lhist[b];
        if (c) atomicAdd(&hist[n * 2048 + b], c);
    }
}

// ---- later passes (keys-cache variant): pure integer histogram over cached keys
__global__ __launch_bounds__(TPB) void k_hist_keys(const unsigned* __restrict__ keys,
                                                   unsigned* __restrict__ hist,
                                                   const unsigned* __restrict__ pfx,
                                                   unsigned shift, unsigned bins, unsigned pmask) {
    __shared__ __align__(16) unsigned tileu[TILE];
    __shared__ unsigned lhist[2048];
    const int n  = blockIdx.y;
    const int t0 = blockIdx.x * TILE;
    for (unsigned b = threadIdx.x; b < bins; b += TPB) lhist[b] = 0u;
    stage_tile(keys + (size_t)n * CHW + t0, tileu);
    const unsigned pv = pfx[n];
    for (int k = 0; k < TILE / TPB; ++k) {
        unsigned key = tileu[threadIdx.x + k * TPB];
        if ((key & pmask) == pv) atomicAdd(&lhist[(key >> shift) & (bins - 1u)], 1u);
    }
    __syncthreads();
    for (unsigned b = threadIdx.x; b < bins; b += TPB) {
        unsigned c = lhist[b];
        if (c) atomicAdd(&hist[n * 2048 + b], c);
    }
}

// ---- recompute-variant histogram (fallback when ws too small for key cache)
__global__ __launch_bounds__(TPB) void k_hist(const float* __restrict__ cls,
                                              const float* __restrict__ sctr,
                                              unsigned* __restrict__ hist,
                                              const unsigned* __restrict__ pfx,
                                              unsigned shift, unsigned bins, unsigned pmask) {
    __shared__ __align__(16) float tile[TILE];
    __shared__ unsigned lhist[2048];
    const int n  = blockIdx.y;
    const int t0 = blockIdx.x * TILE;
    for (unsigned b = threadIdx.x; b < bins; b += TPB) lhist[b] = 0u;
    stage_tile(cls + (size_t)n * CHW + t0, tile);
    const unsigned pv   = pfx[n];
    const float*   ctrb = sctr + n * HWSZ;
    for (int k = 0; k < TILE / TPB; ++k) {
        int   t = threadIdx.x + k * TPB;
        int   l = (t0 + t) % HWSZ;
        float p = sigm(tile[t]);
        float s = (p > 0.05f) ? p * ctrb[l] : -INFINITY;
        unsigned key = keyOf(s);
        if ((key & pmask) == pv) atomicAdd(&lhist[(key >> shift) & (bins - 1u)], 1u);
    }
    __syncthreads();
    for (unsigned b = threadIdx.x; b < bins; b += TPB) {
        unsigned c = lhist[b];
        if (c) atomicAdd(&hist[n * 2048 + b], c);
    }
}

// scan histogram from the top, pick bin containing the TOPK-th key; zero hist for next pass
__global__ void k_scan(unsigned* __restrict__ hist, unsigned* __restrict__ pfx,
                       unsigned* __restrict__ need, unsigned shift, unsigned bins, int init) {
    int n = threadIdx.x;
    if (n >= NIMG) return;
    unsigned* h   = hist + n * 2048;
    unsigned  nd  = init ? (unsigned)TOPK : need[n];
    unsigned  cum = 0;
    for (int b = (int)bins - 1; b >= 0; --b) {
        unsigned c = h[b];
        if (cum + c >= nd) { pfx[n] |= ((unsigned)b) << shift; nd -= cum; break; }
        cum += c;
    }
    need[n] = nd;                       // after last pass: #equal-key entries to accept
    for (unsigned b = 0; b < 2048; ++b) h[b] = 0u;
}

// collect from cached keys: keys > K -> slots [0,G); exactly need[n] keys == K -> slots [G,1000)
__global__ __launch_bounds__(TPB) void k_collect_keys(const unsigned* __restrict__ keys,
                                                      const unsigned* __restrict__ pfx,
                                                      const unsigned* __restrict__ need,
                                                      unsigned* __restrict__ cntHi,
                                                      unsigned* __restrict__ cntEq,
                                                      unsigned long long* __restrict__ pairs) {
    __shared__ __align__(16) unsigned tileu[TILE];
    const int n  = blockIdx.y;
    const int t0 = blockIdx.x * TILE;
    stage_tile(keys + (size_t)n * CHW + t0, tileu);
    const unsigned K  = pfx[n];
    const unsigned ne = need[n];
    for (int k = 0; k < TILE / TPB; ++k) {
        int      t   = threadIdx.x + k * TPB;
        unsigned key = tileu[t];
        if (key >= K) {
            int e = t0 + t;
            int c = e / HWSZ;
            int l = e - c * HWSZ;
            unsigned idxref = (unsigned)(l * NCLS + c);   // reference flat index l*C+c
            unsigned long long pr = ((unsigned long long)key << 32) | (unsigned)(~idxref);
            if (key > K) {
                unsigned slot = atomicAdd(&cntHi[n], 1u);
                if (slot < (unsigned)TOPK) pairs[n * PADK + slot] = pr;
            } else {
                unsigned eidx = atomicAdd(&cntEq[n], 1u);
                if (eidx < ne) pairs[n * PADK + (TOPK - 1) - eidx] = pr;
            }
        }
    }
}

// recompute-variant collect (fallback)
__global__ __launch_bounds__(TPB) void k_collect(const float* __restrict__ cls,
                                                 const float* __restrict__ sctr,
                                                 const unsigned* __restrict__ pfx,
                                                 const unsigned* __restrict__ need,
                                                 unsigned* __restrict__ cntHi,
                                                 unsigned* __restrict__ cntEq,
                                                 unsigned long long* __restrict__ pairs) {
    __shared__ __align__(16) float tile[TILE];
    const int n  = blockIdx.y;
    const int t0 = blockIdx.x * TILE;
    stage_tile(cls + (size_t)n * CHW + t0, tile);
    const unsigned K    = pfx[n];
    const unsigned ne   = need[n];
    const float*   ctrb = sctr + n * HWSZ;
    for (int k = 0; k < TILE / TPB; ++k) {
        int   t = threadIdx.x + k * TPB;
        int   e = t0 + t;
        int   c = e / HWSZ;
        int   l = e - c * HWSZ;
        float p = sigm(tile[t]);
        float s = (p > 0.05f) ? p * ctrb[l] : -INFINITY;
        unsigned key = keyOf(s);
        if (key >= K) {
            unsigned idxref = (unsigned)(l * NCLS + c);
            unsigned long long pr = ((unsigned long long)key << 32) | (unsigned)(~idxref);
            if (key > K) {
                unsigned slot = atomicAdd(&cntHi[n], 1u);
                if (slot < (unsigned)TOPK) pairs[n * PADK + slot] = pr;
            } else {
                unsigned eidx = atomicAdd(&cntEq[n], 1u);
                if (eidx < ne) pairs[n * PADK + (TOPK - 1) - eidx] = pr;
            }
        }
    }
}

// bitonic sort 1024 u64 per image, descending (key desc, tie -> lowest idx first via ~idx)
__global__ __launch_bounds__(1024) void k_sort(unsigned long long* __restrict__ pairs) {
    __shared__ unsigned long long s[PADK];
    const int n = blockIdx.x, t = threadIdx.x;
    s[t] = pairs[n * PADK + t];
    __syncthreads();
    for (int k = 2; k <= PADK; k <<= 1)
        for (int j = k >> 1; j > 0; j >>= 1) {
            int p = t ^ j;
            if (p > t) {
                bool desc = ((t & k) == 0);
                unsigned long long a = s[t], b = s[p];
                if ((a < b) == desc) { s[t] = b; s[p] = a; }
            }
            __syncthreads();
        }
    pairs[n * PADK + t] = s[t];
}

// decode + class-aware greedy NMS (bitmask in LDS) + top-100 output
__global__ __launch_bounds__(1024) void k_nms(const unsigned long long* __restrict__ pairs,
                                              const float* __restrict__ reg,
                                              const float* __restrict__ anch,
                                              float* __restrict__ out) {
    __shared__ unsigned adj[TOPK * 32];   // 128 KB: suppression adjacency bitmask
    __shared__ float    bx[TOPK * 4];
    __shared__ float    ssc[TOPK];
    __shared__ int      scls[TOPK];
    __shared__ unsigned char sval[TOPK];
    __shared__ unsigned supw[32];
    __shared__ int      list[POSTK];
    __shared__ int      cnt;
    const int n = blockIdx.x, t = threadIdx.x;

    // zero this image's slice of d_out (harness poisons it)
    for (int i = t; i < POSTK * 5; i += 1024) out[n * POSTK * 5 + i] = 0.0f;
    for (int i = t; i < POSTK; i += 1024)     out[NIMG * POSTK * 5 + n * POSTK + i] = 0.0f;

    if (t < TOPK) {
        unsigned long long pr = pairs[n * PADK + t];
        unsigned key = (unsigned)(pr >> 32);
        unsigned idx = ~(unsigned)pr;
        float    val = valOf(key);                       // padding(key=0) -> NaN -> invalid
        int loc = (int)(idx / NCLS);
        int c   = (int)(idx - (unsigned)loc * NCLS);
        loc = (loc < HWSZ && loc >= 0) ? loc : 0;        // guard padding garbage
        float ax1 = anch[loc * 4 + 0], ay1 = anch[loc * 4 + 1];
        float ax2 = anch[loc * 4 + 2], ay2 = anch[loc * 4 + 3];
        float wdt = ax2 - ax1 + 1.0f, hgt = ay2 - ay1 + 1.0f;
        float cx  = ax1 + 0.5f * wdt, cy  = ay1 + 0.5f * hgt;
        const float* rb = reg + (size_t)n * 4 * HWSZ;
        float dx = rb[0 * HWSZ + loc] / 10.0f;
        float dy = rb[1 * HWSZ + loc] / 10.0f;
        float dw = fminf(rb[2 * HWSZ + loc] / 5.0f, XCLIP);
        float dh = fminf(rb[3 * HWSZ + loc] / 5.0f, XCLIP);
        float pcx = dx * wdt + cx, pcy = dy * hgt + cy;
        float pw = expf(dw) * wdt, ph = expf(dh) * hgt;
        float x1 = fminf(fmaxf(pcx - 0.5f * (pw - 1.0f), 0.0f), IMGM1);
        float y1 = fminf(fmaxf(pcy - 0.5f * (ph - 1.0f), 0.0f), IMGM1);
        float x2 = fminf(fmaxf(pcx + 0.5f * (pw - 1.0f), 0.0f), IMGM1);
        float y2 = fminf(fmaxf(pcy + 0.5f * (ph - 1.0f), 0.0f), IMGM1);
        bool valid = isfinite(val) && (x2 - x1 + 1.0f >= 0.0f) && (y2 - y1 + 1.0f >= 0.0f);
        bx[t * 4 + 0] = x1; bx[t * 4 + 1] = y1; bx[t * 4 + 2] = x2; bx[t * 4 + 3] = y2;
        ssc[t]  = valid ? sqrtf(val) : 0.0f;
        scls[t] = c + 1;
        sval[t] = valid ? 1 : 0;
    }
    __syncthreads();

    // build adjacency: bit j of row i set iff j>i, same class, IoU > 0.6
    for (int i = t; i < TOPK; i += 1024) {
        float ix1 = bx[i * 4], iy1 = bx[i * 4 + 1], ix2 = bx[i * 4 + 2], iy2 = bx[i * 4 + 3];
        float iar = fmaxf(ix2 - ix1, 0.0f) * fmaxf(iy2 - iy1, 0.0f);
        int   ic  = scls[i];
        for (int w = 0; w < 32; ++w) {
            unsigned bits = 0u;
            int j0 = w * 32;
            for (int b = 0; b < 32; ++b) {
                int j = j0 + b;
                if (j > i && j < TOPK && scls[j] == ic) {
                    float jx1 = bx[j * 4], jy1 = bx[j * 4 + 1], jx2 = bx[j * 4 + 2], jy2 = bx[j * 4 + 3];
                    float jar = fmaxf(jx2 - jx1, 0.0f) * fmaxf(jy2 - jy1, 0.0f);
                    float iw  = fmaxf(fminf(ix2, jx2) - fmaxf(ix1, jx1), 0.0f);
                    float ih  = fmaxf(fminf(iy2, jy2) - fmaxf(iy1, jy1), 0.0f);
                    float inter = iw * ih;
                    float iou = inter / (iar + jar - inter + 1e-9f);
                    if (iou > NMS_T) bits |= (1u << b);
                }
            }
            adj[i * 32 + w] = bits;
        }
    }
    __syncthreads();

    // serial greedy walk on one wave32: lane w owns suppression word w
    if (t < 32) {
        unsigned rem = 0u;
        for (int i = 0; i < TOPK; ++i) {
            unsigned wi = (unsigned)__shfl((int)rem, i >> 5, 32);
            bool supi = (wi >> (i & 31)) & 1u;
            bool keep = sval[i] && !supi;               // uniform across the wave
            if (keep) rem |= adj[i * 32 + t];
        }
        supw[t] = rem;
    }
    __syncthreads();

    if (t == 0) {
        int cc = 0;
        for (int r = 0; r < TOPK && cc < POSTK; ++r) {
            bool sup = (supw[r >> 5] >> (r & 31)) & 1u;
            if (sval[r] && !sup) list[cc++] = r;
        }
        cnt = cc;
    }
    __syncthreads();

    if (t < cnt) {
        int r = list[t];
        float sc = ssc[r];
        if (sc > 0.0f) {                                // ok = topv > 0
            float* o = out + n * POSTK * 5 + t * 5;
            o[0] = bx[r * 4 + 0]; o[1] = bx[r * 4 + 1];
            o[2] = bx[r * 4 + 2]; o[3] = bx[r * 4 + 3]; o[4] = sc;
            out[NIMG * POSTK * 5 + n * POSTK + t] = (float)scls[r];
        }
    }
}

// ---------------- launch ----------------
extern "C" void kernel_launch(void* const* d_in, const int* in_sizes, int n_in,
                              void* d_out, int out_size, void* d_ws, size_t ws_size,
                              hipStream_t stream) {
    (void)in_sizes; (void)n_in; (void)out_size;
    const float* reg  = (const float*)d_in[0];  // [8,4,160,160]
    const float* ctr  = (const float*)d_in[1];  // [8,1,160,160]
    const float* cls  = (const float*)d_in[2];  // [8,80,160,160]
    const float* anch = (const float*)d_in[3];  // [25600,4]
    float* out = (float*)d_out;                 // dets [8,100,5] ++ labels [8,100]
    char*  ws  = (char*)d_ws;

    unsigned* hist  = (unsigned*)(ws + HIST_OFF);
    unsigned* pfx   = (unsigned*)(ws + PFX_OFF);
    unsigned* need  = (unsigned*)(ws + NEED_OFF);
    unsigned* cntHi = (unsigned*)(ws + CNTHI_OFF);
    unsigned* cntEq = (unsigned*)(ws + CNTEQ_OFF);
    unsigned long long* pairs = (unsigned long long*)(ws + PAIRS_OFF);
    float*    sctr  = (float*)(ws + SCTR_OFF);
    unsigned* keys  = (unsigned*)(ws + KEYS_OFF);

    (void)hipMemsetAsync(ws, 0, ZERO_BYTES, stream);  // hists, pfx, need, counters, pairs

    k_ctr<<<(NIMG * HWSZ + 255) / 256, 256, 0, stream>>>(ctr, sctr);

    dim3 g(TILES_PER_IMG, NIMG);
    // exact radix select: bits [31:21], [20:10], [9:0]
    if (ws_size >= WS_BIG) {
        // keys computed once (sigmoid done once), cached; later passes are integer-only
        k_hist_store<<<g, TPB, 0, stream>>>(cls, sctr, keys, hist);
        k_scan<<<1, 32, 0, stream>>>(hist, pfx, need, 21u, 2048u, 1);
        k_hist_keys<<<g, TPB, 0, stream>>>(keys, hist, pfx, 10u, 2048u, 0xFFE00000u);
        k_scan<<<1, 32, 0, stream>>>(hist, pfx, need, 10u, 2048u, 0);
        k_hist_keys<<<g, TPB, 0, stream>>>(keys, hist, pfx, 0u, 1024u, 0xFFFFFC00u);
        k_scan<<<1, 32, 0, stream>>>(hist, pfx, need, 0u, 1024u, 0);
        k_collect_keys<<<g, TPB, 0, stream>>>(keys, pfx, need, cntHi, cntEq, pairs);
    } else {
        // recompute path (no key cache)
        k_hist<<<g, TPB, 0, stream>>>(cls, sctr, hist, pfx, 21u, 2048u, 0u);
        k_scan<<<1, 32, 0, stream>>>(hist, pfx, need, 21u, 2048u, 1);
        k_hist<<<g, TPB, 0, stream>>>(cls, sctr, hist, pfx, 10u, 2048u, 0xFFE00000u);
        k_scan<<<1, 32, 0, stream>>>(hist, pfx, need, 10u, 2048u, 0);
        k_hist<<<g, TPB, 0, stream>>>(cls, sctr, hist, pfx, 0u, 1024u, 0xFFFFFC00u);
        k_scan<<<1, 32, 0, stream>>>(hist, pfx, need, 0u, 1024u, 0);
        k_collect<<<g, TPB, 0, stream>>>(cls, sctr, pfx, need, cntHi, cntEq, pairs);
    }

    k_sort<<<NIMG, 1024, 0, stream>>>(pairs);
    k_nms<<<NIMG, 1024, 0, stream>>>(pairs, reg, anch, out);
}